// GINNet_67336497266911
// MI455X (gfx1250) — compile-verified
//
#include <hip/hip_runtime.h>
#include <hip/hip_bf16.h>

typedef __attribute__((ext_vector_type(2))) float v2f;
typedef __attribute__((ext_vector_type(8))) float v8f;

// ---------------------------------------------------------------------------
// Vectorized copy: dst = src (used for the GIN self-term: agg = x)
// ---------------------------------------------------------------------------
__global__ void copy_f4(const float4* __restrict__ src, float4* __restrict__ dst,
                        long n4) {
    long i = (long)blockIdx.x * blockDim.x + threadIdx.x;
    if (i < n4) dst[i] = src[i];
}

// ---------------------------------------------------------------------------
// Edge scatter-add: agg[dst] += x[src]; one thread per (edge, float4-chunk).
// D4 = feature_dim / 4 (compile-time so /,% become shifts/masks).
// Atomic f32 adds resolve in L2 (agg fits the 192MB L2 on MI455X).
// ---------------------------------------------------------------------------
template <int D4>
__global__ void scatter_add_k(const float4* __restrict__ x,
                              const int* __restrict__ sidx,
                              const int* __restrict__ didx,
                              float* __restrict__ agg, long total) {
    long i = (long)blockIdx.x * blockDim.x + threadIdx.x;
    if (i >= total) return;
    int e = (int)(i / D4);
    int q = (int)(i % D4);
    int s = sidx[e];
    int d = didx[e];
    float4 v = x[(long)s * D4 + q];
    float* p = agg + ((long)d * D4 + q) * 4;
    atomicAdd(p + 0, v.x);
    atomicAdd(p + 1, v.y);
    atomicAdd(p + 2, v.z);
    atomicAdd(p + 3, v.w);
}

// ---------------------------------------------------------------------------
// f32 WMMA GEMM: C = act(A(MxK) @ W(KxN) + bias), act per MODE.
// MODE: 0 = identity, 1 = ReLU, 2 = ELU(alpha=1).
// One wave (32 lanes) computes a 16x64 output tile: 4 x v8f accumulators,
// K-loop in steps of 4 via V_WMMA_F32_16X16X4_F32 (full fp32 precision).
//
// Fragment layouts (CDNA5 ISA 7.12.2):
//   A 16x4 f32 : lanes 0-15 -> M=lane,  v0=K0, v1=K1
//                lanes16-31 -> M=lane-16,v0=K2, v1=K3
//   B 4x16 f32 : lanes 0-15 -> N=lane,  v0=K0, v1=K1
//                lanes16-31 -> N=lane-16,v0=K2, v1=K3
//   C 16x16 f32: vr, lanes 0-15 -> M=r, N=lane ; lanes 16-31 -> M=r+8.
// ---------------------------------------------------------------------------
template <int MODE>
__global__ __launch_bounds__(256) void gemm_wmma_f32(
    const float* __restrict__ A, const float* __restrict__ W,
    const float* __restrict__ bias, float* __restrict__ C,
    int M, int K, int Nn) {
    const int lane    = threadIdx.x & 31;
    const int waveId  = threadIdx.x >> 5;
    const int tilesN  = Nn >> 6;                 // 64 output cols per wave
    const int tileIdx = blockIdx.x * 8 + waveId; // 8 waves per block
    const int tm = tileIdx / tilesN;
    const int tn = tileIdx % tilesN;
    if (tm * 16 >= M) return;                    // wave-uniform exit

    const int halfSel = lane >> 4;   // 0: lanes 0-15, 1: lanes 16-31
    const int l15     = lane & 15;

    const int row    = tm * 16 + l15;
    const int rowC   = row < M ? row : (M - 1);  // clamp: loads stay in-bounds
    const float amsk = row < M ? 1.0f : 0.0f;    // zero contribution if padded
    const int colB   = tn * 64 + l15;

    v8f acc0 = {}, acc1 = {}, acc2 = {}, acc3 = {};

    for (int k = 0; k < K; k += 4) {
        const int kk = k + halfSel * 2;

        v2f a;
        a.x = A[(long)rowC * K + kk] * amsk;
        a.y = A[(long)rowC * K + kk + 1] * amsk;

        const float* Wr0 = W + (long)kk * Nn + colB;
        const float* Wr1 = W + (long)(kk + 1) * Nn + colB;
        v2f b0, b1, b2, b3;
        b0.x = Wr0[0];  b0.y = Wr1[0];
        b1.x = Wr0[16]; b1.y = Wr1[16];
        b2.x = Wr0[32]; b2.y = Wr1[32];
        b3.x = Wr0[48]; b3.y = Wr1[48];

        acc0 = __builtin_amdgcn_wmma_f32_16x16x4_f32(false, a, false, b0,
                                                     (short)0, acc0, false, false);
        acc1 = __builtin_amdgcn_wmma_f32_16x16x4_f32(false, a, false, b1,
                                                     (short)0, acc1, false, false);
        acc2 = __builtin_amdgcn_wmma_f32_16x16x4_f32(false, a, false, b2,
                                                     (short)0, acc2, false, false);
        acc3 = __builtin_amdgcn_wmma_f32_16x16x4_f32(false, a, false, b3,
                                                     (short)0, acc3, false, false);
    }

    v8f accs[4] = {acc0, acc1, acc2, acc3};
#pragma unroll
    for (int j = 0; j < 4; ++j) {
        const int col = tn * 64 + j * 16 + l15;
        const float bv = bias[col];
#pragma unroll
        for (int r = 0; r < 8; ++r) {
            const int orow = tm * 16 + r + halfSel * 8;
            if (orow < M) {
                float v = accs[j][r] + bv;
                if (MODE == 1) v = fmaxf(v, 0.0f);
                if (MODE == 2) v = v > 0.0f ? v : (__expf(v) - 1.0f);
                C[(long)orow * Nn + col] = v;
            }
        }
    }
}

// ---------------------------------------------------------------------------
// Host-side orchestration (graph-capture safe: launches only).
// ---------------------------------------------------------------------------
extern "C" void kernel_launch(void* const* d_in, const int* in_sizes, int n_in,
                              void* d_out, int out_size, void* d_ws, size_t ws_size,
                              hipStream_t stream) {
    const float* feature = (const float*)d_in[0];
    const int*   edge    = (const int*)d_in[1];
    const float* w1 = (const float*)d_in[2];
    const float* b1 = (const float*)d_in[3];
    const float* w2 = (const float*)d_in[4];
    const float* b2 = (const float*)d_in[5];
    const float* w3 = (const float*)d_in[6];
    const float* b3 = (const float*)d_in[7];
    const float* w4 = (const float*)d_in[8];
    const float* b4 = (const float*)d_in[9];
    float* out = (float*)d_out;

    const int D_IN = 128, H = 512;
    const int N = in_sizes[0] / D_IN;
    const int E = in_sizes[1] / 2;

    const int* srcI = edge;       // edge_index[0] = source nodes
    const int* dstI = edge + E;   // edge_index[1] = destination nodes

    // Workspace layout (all float, 16B-aligned offsets)
    float* agg1 = (float*)d_ws;                  // N x 128
    float* t1   = agg1 + (size_t)N * D_IN;       // N x 512 (also reused as t2)
    float* h    = t1   + (size_t)N * H;          // N x 512
    float* agg2 = h    + (size_t)N * H;          // N x 512
    float* t2   = t1;

    const int TPB = 256;
    auto gemmBlocks = [](int M, int Nn) {
        long tiles = (long)((M + 15) / 16) * (Nn >> 6);
        return (int)((tiles + 7) / 8);
    };

    // ---- Layer 1 aggregation: agg1 = x + scatter_add(x[src] -> dst) ----
    {
        long n4 = (long)N * D_IN / 4;
        copy_f4<<<(int)((n4 + TPB - 1) / TPB), TPB, 0, stream>>>(
            (const float4*)feature, (float4*)agg1, n4);
        long total = (long)E * (D_IN / 4);
        scatter_add_k<D_IN / 4><<<(int)((total + TPB - 1) / TPB), TPB, 0, stream>>>(
            (const float4*)feature, srcI, dstI, agg1, total);
    }

    // ---- conv1 MLP: t1 = relu(agg1 @ w1 + b1); h = elu(t1 @ w2 + b2) ----
    gemm_wmma_f32<1><<<gemmBlocks(N, H), TPB, 0, stream>>>(agg1, w1, b1, t1, N, D_IN, H);
    gemm_wmma_f32<2><<<gemmBlocks(N, H), TPB, 0, stream>>>(t1, w2, b2, h, N, H, H);

    // ---- Layer 2 aggregation: agg2 = h + scatter_add(h[src] -> dst) ----
    {
        long n4 = (long)N * H / 4;
        copy_f4<<<(int)((n4 + TPB - 1) / TPB), TPB, 0, stream>>>(
            (const float4*)h, (float4*)agg2, n4);
        long total = (long)E * (H / 4);
        scatter_add_k<H / 4><<<(int)((total + TPB - 1) / TPB), TPB, 0, stream>>>(
            (const float4*)h, srcI, dstI, agg2, total);
    }

    // ---- conv2 MLP: t2 = relu(agg2 @ w3 + b3); out = t2 @ w4 + b4 ----
    gemm_wmma_f32<1><<<gemmBlocks(N, H), TPB, 0, stream>>>(agg2, w3, b3, t2, N, H, H);
    gemm_wmma_f32<0><<<gemmBlocks(N, D_IN), TPB, 0, stream>>>(t2, w4, b4, out, N, H, D_IN);
}